// Sampler_41832981463502
// MI455X (gfx1250) — compile-verified
//
#include <hip/hip_runtime.h>
#include <hip/hip_bf16.h>
#include <stdint.h>

#define IGNORED_LOGITS (-3000.0f)
#define TOPN   256
#define CAP    1024
#define NBINS  4096
#define CHUNK  4000
#define K1T    256

typedef unsigned int uint32x4 __attribute__((ext_vector_type(4)));
typedef int          int32x8  __attribute__((ext_vector_type(8)));
typedef int          int32x4  __attribute__((ext_vector_type(4)));
typedef float        v2f      __attribute__((ext_vector_type(2)));
typedef float        v8f      __attribute__((ext_vector_type(8)));

// order-preserving f32 -> u32 key (larger key == larger float)
__device__ __forceinline__ unsigned orderKey(float x) {
  unsigned b = __float_as_uint(x);
  return (b & 0x80000000u) ? ~b : (b | 0x80000000u);
}
__device__ __forceinline__ float unKey(unsigned u) {
  return __uint_as_float((u & 0x80000000u) ? (u & 0x7FFFFFFFu) : ~u);
}

// ---------------------------------------------------------------------------
// TDM: issue an async 1-D tile load (nelts f32) from global -> LDS.
// D# built per CDNA5 ISA (group0: count=1,type=2;
// group1: data_size=4B, tensor==tile so no OOB edge).
// 6-arg builtin form (clang-23 / therock headers).
// ---------------------------------------------------------------------------
__device__ __forceinline__ void tdm_issue(unsigned long long gaddr, int nelts,
                                          unsigned ldsOff) {
  uint32x4 g0;
  g0[0] = 1u;                                            // count=1, user D#
  g0[1] = ldsOff;                                        // LDS byte address
  g0[2] = (unsigned)(gaddr & 0xFFFFFFFFull);             // global_addr lo
  g0[3] = (unsigned)((gaddr >> 32) & 0x01FFFFFFull) | (2u << 30); // hi | type=2
  int32x8 g1;
  g1[0] = (int)(2u << 16);                               // data_size=4B
  g1[1] = (int)(((unsigned)nelts & 0xFFFFu) << 16);      // tensor_dim0 lo16
  g1[2] = (int)((((unsigned)nelts >> 16) & 0xFFFFu) | (1u << 16)); // dim0 hi | tensor_dim1=1
  g1[3] = (int)(((unsigned)nelts & 0xFFFFu) << 16);      // tile_dim0
  g1[4] = 1;                                             // tile_dim1=1, tile_dim2=0
  g1[5] = nelts;                                         // tensor_dim0_stride lo
  g1[6] = 0;
  g1[7] = 0;
  int32x4 z4 = {0, 0, 0, 0};
  int32x8 z8 = {0, 0, 0, 0, 0, 0, 0, 0};
  __builtin_amdgcn_tensor_load_to_lds(g0, g1, z4, z4, z8, 0);
}

// ---------------------------------------------------------------------------
// K0: init global min(first-prob) to +inf bits
// ---------------------------------------------------------------------------
__global__ void init_kernel(unsigned* gminBits) {
  if (threadIdx.x == 0) *gminBits = 0x7F800000u;  // +inf
}

// ---------------------------------------------------------------------------
// K1: per-row top-256 selection (TDM-streamed histogram + bitonic sort),
//     plus per-row p0 = 1/sum(exp((v_i - v_0)/T), i<k)  -> global atomicMin.
// ---------------------------------------------------------------------------
__global__ __launch_bounds__(K1T)
void topk_select_kernel(const float* __restrict__ logits,
                        const float* __restrict__ params,
                        float* __restrict__ outVals, int* __restrict__ outIdx,
                        unsigned* __restrict__ gminBits, int V) {
  __shared__ int      hist[NBINS];
  __shared__ unsigned ckey[CAP];
  __shared__ int      cidx[CAP];
  __shared__ float    stage[2][CHUNK];
  __shared__ float    red[K1T];
  __shared__ int      iscan[K1T];
  __shared__ int      scnt;
  __shared__ int      bstar;

  const int row = blockIdx.x;
  const int tid = threadIdx.x;
  const float* rowp = logits + (size_t)row * (size_t)V;

  for (int i = tid; i < NBINS; i += K1T) hist[i] = 0;
  if (tid == 0) { scnt = 0; bstar = 0; }
  __syncthreads();

  // ---- Pass 1: histogram of order keys, streamed via Tensor Data Mover ----
  const int nChunk = (V + CHUNK - 1) / CHUNK;
  const unsigned long long rowAddr = (unsigned long long)(uintptr_t)rowp;
  const bool issuer = (tid < 32);  // wave 0 owns TENSORcnt

  if (issuer)
    tdm_issue(rowAddr, min(CHUNK, V), (unsigned)(uintptr_t)&stage[0][0]);
  for (int c = 0; c < nChunk; ++c) {
    if (issuer) __builtin_amdgcn_s_wait_tensorcnt(0);  // chunk c landed
    __syncthreads();
    if (issuer && (c + 1) < nChunk) {                  // prefetch chunk c+1
      int nb = (c + 1) * CHUNK;
      tdm_issue(rowAddr + (unsigned long long)nb * 4ull, min(CHUNK, V - nb),
                (unsigned)(uintptr_t)&stage[(c + 1) & 1][0]);
    }
    const int cs = min(CHUNK, V - c * CHUNK);
    const float* buf = stage[c & 1];
    for (int i = tid; i < cs; i += K1T) {
      unsigned u = orderKey(buf[i]);
      atomicAdd(&hist[u >> 20], 1);
    }
    __syncthreads();  // done reading buf before it is refilled next+1 iter
  }

  // ---- find threshold bin: largest b* with count(bins >= b*) >= 256 ----
  const int segTop = NBINS - tid * 16;  // thread tid owns [segTop-16, segTop)
  int s = 0;
  for (int b = segTop - 16; b < segTop; ++b) s += hist[b];
  iscan[tid] = s;
  __syncthreads();
  for (int off = 1; off < K1T; off <<= 1) {
    int v = (tid >= off) ? iscan[tid - off] : 0;
    __syncthreads();
    iscan[tid] += v;
    __syncthreads();
  }
  int acc = iscan[tid] - s;  // count strictly above this segment
  for (int b = segTop - 1; b >= segTop - 16; --b) {
    int nb = acc + hist[b];
    if (acc < TOPN && nb >= TOPN) bstar = b;  // unique crossing writer
    acc = nb;
  }
  __syncthreads();
  const int thr = bstar;

  // ---- Pass 2: gather candidates (L2-resident re-read) ----
  for (int i = tid; i < V; i += K1T) {
    unsigned u = orderKey(rowp[i]);
    if ((int)(u >> 20) >= thr) {
      int pos = atomicAdd(&scnt, 1);
      if (pos < CAP) { ckey[pos] = u; cidx[pos] = i; }
    }
  }
  __syncthreads();
  int cnt = scnt; if (cnt > CAP) cnt = CAP;
  for (int i = tid; i < CAP; i += K1T)
    if (i >= cnt) { ckey[i] = 0u; cidx[i] = 0; }
  __syncthreads();

  // ---- bitonic sort, descending by key ----
  for (unsigned k = 2; k <= CAP; k <<= 1) {
    for (unsigned j = k >> 1; j > 0; j >>= 1) {
      for (unsigned i = tid; i < CAP; i += K1T) {
        unsigned ixj = i ^ j;
        if (ixj > i) {
          unsigned a = ckey[i], b = ckey[ixj];
          bool up = ((i & k) == 0);
          if (up ? (a < b) : (a > b)) {
            ckey[i] = b; ckey[ixj] = a;
            int t = cidx[i]; cidx[i] = cidx[ixj]; cidx[ixj] = t;
          }
        }
      }
      __syncthreads();
    }
  }

  // ---- emit sorted top-256 and per-row first softmax prob p0 ----
  const float v0   = unKey(ckey[0]);
  const int   kk   = (int)params[row * 3 + 0];
  const float temp = params[row * 3 + 2];
  float e = 0.0f;
  if (tid < TOPN) {
    float v = unKey(ckey[tid]);
    outVals[row * TOPN + tid] = v;
    outIdx [row * TOPN + tid] = cidx[tid];
    if (tid < kk) e = __expf((v - v0) / temp);
  }
  red[tid] = e;
  __syncthreads();
  for (int off = K1T / 2; off > 0; off >>= 1) {
    if (tid < off) red[tid] += red[tid + off];
    __syncthreads();
  }
  if (tid == 0) {
    float p0 = 1.0f / red[0];           // cum[row][0] of first softmax
    atomicMin(gminBits, __float_as_uint(p0));  // positive floats: bit order == value order
  }
}

// ---------------------------------------------------------------------------
// WMMA-based 256-wide inclusive prefix sum (one wave32).
// X[r][c] = sp[16r+c].  P = X*U (U upper-tri ones); cum = Lstrict*S + P.
// C/D layout: vgpr j, lane -> row r = j + 8*(lane>=16), col n = lane&15.
// ---------------------------------------------------------------------------
__device__ __forceinline__ v8f wmma4(v2f a, v2f b, v8f c) {
  return __builtin_amdgcn_wmma_f32_16x16x4_f32(false, a, false, b,
                                               (short)0, c, false, false);
}

__device__ __forceinline__ v8f wave_cumsum256(const float* sp, float* sseg,
                                              int half, int n) {
  v8f P = {};
  #pragma unroll
  for (int kb = 0; kb < 4; ++kb) {
    int k0 = 4 * kb + 2 * half;
    v2f a; a.x = sp[16 * n + k0];            a.y = sp[16 * n + k0 + 1];
    v2f b; b.x = (k0     <= n) ? 1.f : 0.f;  b.y = (k0 + 1 <= n) ? 1.f : 0.f;
    P = wmma4(a, b, P);
  }
  if (n == 15) {                              // row totals P[r][15]
    #pragma unroll
    for (int j = 0; j < 8; ++j) sseg[j + 8 * half] = P[j];
  }
  __syncthreads();
  v8f C = P;
  #pragma unroll
  for (int kb = 0; kb < 4; ++kb) {
    int k0 = 4 * kb + 2 * half;
    v2f a; a.x = (k0 < n) ? 1.f : 0.f;  a.y = (k0 + 1 < n) ? 1.f : 0.f;
    v2f b; b.x = sseg[k0];              b.y = sseg[k0 + 1];
    C = wmma4(a, b, C);
  }
  __syncthreads();
  return C;
}

// ---------------------------------------------------------------------------
// K2: per-row sampling over the sorted top-256 (1 wave per row).
// ---------------------------------------------------------------------------
__global__ __launch_bounds__(32)
void sample_kernel(const float* __restrict__ tvals, const int* __restrict__ tidx,
                   const float* __restrict__ params,
                   const unsigned* __restrict__ gminBits,
                   int* __restrict__ outTok) {
  __shared__ float sv[TOPN];   // masked/temp-scaled logits
  __shared__ float sp[TOPN];   // probs staging (WMMA A operand source)
  __shared__ int   si[TOPN];
  __shared__ float sseg[16];

  const int row  = blockIdx.x;
  const int lane = threadIdx.x;
  const int half = lane >> 4;
  const int n    = lane & 15;

  const int   kk   = (int)params[row * 3 + 0];
  const float tp0  = params[row * 3 + 1];
  const float temp = params[row * 3 + 2];

  for (int e = lane; e < TOPN; e += 32) {
    float v = tvals[row * TOPN + e];
    si[e] = tidx[row * TOPN + e];
    sv[e] = ((e >= kk) ? IGNORED_LOGITS : v) / temp;   // top-k mask, then /T
  }
  __syncthreads();
  const float m = sv[0];  // row max (k>=1 so element 0 never masked)

  // softmax #1
  float lsum = 0.f;
  for (int e = lane; e < TOPN; e += 32) {
    float ex = __expf(sv[e] - m); sp[e] = ex; lsum += ex;
  }
  for (int off = 16; off; off >>= 1) lsum += __shfl_xor(lsum, off, 32);
  __syncthreads();
  const float inv1 = 1.0f / lsum;
  for (int e = lane; e < TOPN; e += 32) sp[e] *= inv1;
  __syncthreads();

  v8f cum1 = wave_cumsum256(sp, sseg, half, n);        // WMMA cumsum #1

  const float gmin = __uint_as_float(*gminBits);       // jnp.min(cum) global
  const float tp   = fmaxf(gmin, tp0);

  // top-p mask -> softmax #2 (keep element 0 so the distribution stays valid)
  float lsum2 = 0.f;
  float e2[8];
  #pragma unroll
  for (int j = 0; j < 8; ++j) {
    int r = j + 8 * half, e = 16 * r + n;
    bool masked = (cum1[j] > tp) && (e != 0);
    float v2 = masked ? IGNORED_LOGITS : sv[e];
    e2[j] = __expf(v2 - m);
    lsum2 += e2[j];
  }
  for (int off = 16; off; off >>= 1) lsum2 += __shfl_xor(lsum2, off, 32);
  const float inv2 = 1.0f / lsum2;
  __syncthreads();
  #pragma unroll
  for (int j = 0; j < 8; ++j) {
    int r = j + 8 * half, e = 16 * r + n;
    sp[e] = e2[j] * inv2;
  }
  __syncthreads();

  v8f cum2 = wave_cumsum256(sp, sseg, half, n);        // WMMA cumsum #2

  // deterministic per-row uniform (stand-in for jax.random.uniform(key(42)))
  unsigned long long z = 42ull + 0x9E3779B97F4A7C15ull * (unsigned long long)(row + 1);
  z += 0x9E3779B97F4A7C15ull;
  z = (z ^ (z >> 30)) * 0xBF58476D1CE4E5B9ull;
  z = (z ^ (z >> 27)) * 0x94D049BB133111EBull;
  z ^= z >> 31;
  const float rnd = (float)(unsigned)(z >> 40) * (1.0f / 16777216.0f);

  int cnt = 0;
  #pragma unroll
  for (int j = 0; j < 8; ++j) cnt += (rnd > cum2[j]) ? 1 : 0;
  for (int off = 16; off; off >>= 1) cnt += __shfl_xor(cnt, off, 32);
  if (cnt > TOPN - 1) cnt = TOPN - 1;
  if (lane == 0) outTok[row] = si[cnt];
}

// ---------------------------------------------------------------------------
extern "C" void kernel_launch(void* const* d_in, const int* in_sizes, int n_in,
                              void* d_out, int out_size, void* d_ws, size_t ws_size,
                              hipStream_t stream) {
  const float* logits = (const float*)d_in[0];
  const float* params = (const float*)d_in[1];
  const int B = in_sizes[1] / 3;
  const int V = in_sizes[0] / B;

  unsigned char* ws = (unsigned char*)d_ws;
  unsigned* gmin  = (unsigned*)ws;
  float*    tvals = (float*)(ws + 256);
  int*      tidx  = (int*)(ws + 256 + (size_t)B * TOPN * sizeof(float));

  init_kernel<<<1, 1, 0, stream>>>(gmin);
  topk_select_kernel<<<B, K1T, 0, stream>>>(logits, params, tvals, tidx, gmin, V);
  sample_kernel<<<B, 32, 0, stream>>>(tvals, tidx, params, gmin, (int*)d_out);
}